// Wavelet_cnn_52166672777892
// MI455X (gfx1250) — compile-verified
//
#include <hip/hip_runtime.h>

typedef float v2f __attribute__((ext_vector_type(2)));
typedef float v4f __attribute__((ext_vector_type(4)));
typedef float v8f __attribute__((ext_vector_type(8)));

// ---------------------------------------------------------------------------
// Kernel 1: one wavelet cascade level via V_WMMA_F32_16X16X4_F32.
// out[t,e] = sum_k w[k] * x[2t + k - 31, e]   (causal, stride 2, L=32)
// One 128-thread block (4 waves) per (batch, 16-row t-tile); each wave owns a
// 16-electrode slice.  C(16x16) = W(16x64) @ X(64x16), W[r,j] = w[j-2r].
// X window (64 rows x 64 e = 16 KB) staged in LDS via coalesced b128 loads
// (clamp + mask-multiply: no exec-mask branches).  Filter taps staged as
// zero-padded 96-float LDS arrays so fragment reads are unconditional ds
// loads.  16 K-steps x 2 filters = 32 chained WMMAs per wave.
// ---------------------------------------------------------------------------
__global__ __launch_bounds__(128) void wavelet_level_wmma(
    const float* __restrict__ xin,   // (B, T_in, 64)
    float* __restrict__ a_out,       // (B, T_out, 64)  -> a_coef slot in d_out
    float* __restrict__ b_out,       // (B, T_out, 64)  -> next-level b, or null
    const float* __restrict__ l_w,   // 32 taps
    const float* __restrict__ h_w,   // 32 taps
    const float* __restrict__ ld_w,  // scalar
    const float* __restrict__ hd_w,  // scalar
    int T_in, int T_out, int numTileT)
{
    __shared__ float lds_x[64 * 64];   // window rows j=0..63, 64 electrodes
    __shared__ float lds_hw[96];       // h taps at [tap+32], zero elsewhere
    __shared__ float lds_lw[96];       // l taps at [tap+32], zero elsewhere

    const int tid = threadIdx.x;
    const int tt  = blockIdx.x % numTileT;           // seq tile
    const int bb  = blockIdx.x / numTileT;           // batch
    const int t0  = tt * 16;
    const int base = 2 * t0 - 31;                    // seq row of window j=0
    const float* xb = xin + (long)bb * T_in * 64;

    // Stage zero-padded filter rows: valid taps live at index tap+32.
    for (int p = tid; p < 96; p += 128) {
        const bool valid = (p >= 32) && (p < 64);
        const int  tap   = valid ? (p - 32) : 0;
        const float m    = valid ? 1.0f : 0.0f;
        lds_hw[p] = m * h_w[tap];
        lds_lw[p] = m * l_w[tap];
    }

    // Stage X window: 64 rows x 16 v4f = 1024 v4f, 8 per thread, coalesced.
    for (int q = tid; q < 1024; q += 128) {
        const int row = q >> 4;                      // 0..63
        const int c4  = q & 15;                      // v4f slot within row
        const int j   = base + row;
        int jc = j < 0 ? 0 : j;                      // clamp to valid row
        jc = jc >= T_in ? (T_in - 1) : jc;
        const float m = (j >= 0 && j < T_in) ? 1.0f : 0.0f;
        v4f val = ((const v4f*)(xb + (long)jc * 64))[c4];
        ((v4f*)lds_x)[(row << 4) + c4] = m * val;
    }
    __syncthreads();

    const int  lane = tid & 31;
    const int  e0   = (tid >> 5) * 16;               // wave's electrode slice
    const int  r    = lane & 15;                     // A row / B,C column
    const bool hi   = lane >= 16;
    const int  c0   = hi ? 2 : 0;                    // K sub-slot within step

    v8f acc_h = {0.f,0.f,0.f,0.f,0.f,0.f,0.f,0.f};
    v8f acc_l = {0.f,0.f,0.f,0.f,0.f,0.f,0.f,0.f};

#pragma unroll
    for (int kk = 0; kk < 16; ++kk) {
        const int j0 = 4 * kk + c0;                  // LDS window row
        v2f bf = { lds_x[(j0 << 6) + e0 + r],
                   lds_x[((j0 + 1) << 6) + e0 + r] };
        const int p0 = 4 * kk + c0 - 2 * r + 32;     // padded tap index [2,95]
        v2f ah = { lds_hw[p0], lds_hw[p0 + 1] };
        v2f al = { lds_lw[p0], lds_lw[p0 + 1] };

        acc_h = __builtin_amdgcn_wmma_f32_16x16x4_f32(
            false, ah, false, bf, (short)0, acc_h, false, false);
        acc_l = __builtin_amdgcn_wmma_f32_16x16x4_f32(
            false, al, false, bf, (short)0, acc_l, false, false);
    }

    const float hd = hd_w[0];
    const float ld = ld_w[0];

    // C layout: VGPR v -> row m = v + (hi ? 8 : 0), column = lane & 15.
#pragma unroll
    for (int v = 0; v < 8; ++v) {
        const int t = t0 + v + (hi ? 8 : 0);
        if (t < T_out) {
            const long idx = ((long)bb * T_out + t) * 64 + e0 + r;
            a_out[idx] = hd * acc_h[v];
            if (b_out) b_out[idx] = ld * acc_l[v];
        }
    }
}

// ---------------------------------------------------------------------------
// Kernel 2: expand a_coef -> scalegram + decomposition (pure streaming).
// scalegram[b,lvl,s,e] = a_coef[lvl][b, s>>(lvl+1), e] * prod_j up_w[bit_j(s)]
// decomposition = sum over lvl.  ~1 GB of nontemporal stores; a_coef reads
// (67 MB total) hit L2.  v4f along the contiguous electrode axis.
// ---------------------------------------------------------------------------
__global__ __launch_bounds__(256) void wavelet_expand(
    const float* __restrict__ coefs,  // a_coef[0..12] packed (d_out + 14*D)
    const float* __restrict__ up_w,   // 2 taps
    float* __restrict__ decomp,       // (B, 8192, 64)
    float* __restrict__ scal)         // (B, 13, 8192, 64)
{
    const long gid = (long)blockIdx.x * 256 + threadIdx.x; // 32*8192*16 total
    const int e4 = (int)(gid & 15);          // v4f index within 64 floats
    const int s  = (int)((gid >> 4) & 8191);
    const int bb = (int)(gid >> 17);

    const float u0 = up_w[0];
    const float u1 = up_w[1];

    v4f   acc = {0.f, 0.f, 0.f, 0.f};
    float f   = 1.0f;
    long  coefOff = 0;
    int   T   = 4096;

#pragma unroll
    for (int lvl = 0; lvl < 13; ++lvl) {
        f *= ((s >> lvl) & 1) ? u1 : u0;
        const int t = s >> (lvl + 1);
        const v4f* src =
            (const v4f*)(coefs + coefOff + ((long)bb * T + t) * 64);
        v4f o = f * src[e4];
        acc += o;

        v4f* dst =
            (v4f*)(scal + (((long)bb * 13 + lvl) * 8192 + s) * 64);
        __builtin_nontemporal_store(o, dst + e4);

        coefOff += 32L * T * 64;   // B * T_lvl * E elements per level
        T >>= 1;
    }

    v4f* dd = (v4f*)(decomp + ((long)bb * 8192 + s) * 64);
    __builtin_nontemporal_store(acc, dd + e4);
}

// ---------------------------------------------------------------------------
// Host side.  Inputs (setup_inputs order): x, l_w(32), h_w(32), ld_w(1),
// hd_w(1), up_w(2).  Output layout (flat, return order): decomposition (D),
// scalegram (13*D), a_coef[0..12] (sum_i 32*64*(4096>>i)).  D = 32*8192*64.
// Workspace: ping-pong b-cascade buffers (33.5 MB + 16.8 MB).
// ---------------------------------------------------------------------------
extern "C" void kernel_launch(void* const* d_in, const int* in_sizes, int n_in,
                              void* d_out, int out_size, void* d_ws, size_t ws_size,
                              hipStream_t stream) {
    (void)in_sizes; (void)n_in; (void)out_size; (void)ws_size;

    const float* x    = (const float*)d_in[0];
    const float* l_w  = (const float*)d_in[1];
    const float* h_w  = (const float*)d_in[2];
    const float* ld_w = (const float*)d_in[3];
    const float* hd_w = (const float*)d_in[4];
    const float* up_w = (const float*)d_in[5];

    float* out = (float*)d_out;
    const long D = 32L * 8192L * 64L;          // 16,777,216
    float* decomp = out;
    float* scal   = out + D;
    float* coefs  = out + 14L * D;             // packed a_coef[0..12]

    // Ping-pong workspace for the b cascade.
    float* ws0 = (float*)d_ws;                 // holds b_1, b_3, b_5, ...
    float* ws1 = ws0 + 32L * 4096L * 64L;      // holds b_2, b_4, ...

    long coefOff = 0;
    for (int lvl = 0; lvl < 13; ++lvl) {
        const int T_in  = 8192 >> lvl;
        const int T_out = T_in >> 1;
        const int numTileT = (T_out + 15) / 16;

        const float* in = (lvl == 0) ? x : ((lvl & 1) ? ws0 : ws1);
        float* a_out = coefs + coefOff;
        float* b_out = (lvl < 12) ? ((lvl & 1) ? ws1 : ws0) : nullptr;

        const int grid = 32 * numTileT;        // B * t-tiles (4 waves/block)
        wavelet_level_wmma<<<grid, 128, 0, stream>>>(
            in, a_out, b_out, l_w, h_w, ld_w, hd_w, T_in, T_out, numTileT);

        coefOff += 32L * T_out * 64L;
    }

    // Expand: 32*8192*16 threads, 16384 blocks of 256.
    wavelet_expand<<<16384, 256, 0, stream>>>(coefs, up_w, decomp, scal);
}